// PropertyEncoderRepresentation_50663434224017
// MI455X (gfx1250) — compile-verified
//
#include <hip/hip_runtime.h>

#define BATCH   16384
#define IN_DIM  768
#define DIM     1024
#define ROWS    32          // M tile per block (two 16-row WMMA tiles per wave)
#define KCHUNK  384         // K staged in 2 chunks to fit static LDS
#define LDSS    388         // 384 + 4 floats pad -> rows spread over LDS banks

typedef __attribute__((ext_vector_type(2))) float v2f;
typedef __attribute__((ext_vector_type(8))) float v8f;

__global__ void zero_counts_kernel(int* counts) {
    if (threadIdx.x < 4) counts[threadIdx.x] = 0;
}

// Compact batch rows by entity type. Output is order-independent (each row's
// result depends only on its own data), so atomic ordering nondeterminism is fine.
__global__ void route_kernel(const int* __restrict__ indices,
                             const int* __restrict__ etypes,
                             const int* __restrict__ edidx,
                             int* __restrict__ counts,
                             int* __restrict__ rowlist,
                             int* __restrict__ didxlist) {
    int i = blockIdx.x * blockDim.x + threadIdx.x;
    if (i >= BATCH) return;
    int ent = indices[i];
    int t = etypes[ent];
    int pos = atomicAdd(&counts[t], 1);
    rowlist[t * BATCH + pos] = i;
    if (t < 3) didxlist[t * BATCH + pos] = edidx[ent];
}

// Type-3 rows: out[i] = lookup[indices[i]]  (1024 f32 = 256 float4 per row)
__global__ void lookup_copy_kernel(const int* __restrict__ counts,
                                   const int* __restrict__ rowlist,
                                   const int* __restrict__ indices,
                                   const float* __restrict__ lookup,
                                   float* __restrict__ out) {
    int b = blockIdx.x;
    if (b >= counts[3]) return;
    int i   = rowlist[3 * BATCH + b];
    int ent = indices[i];
    const float4* src = (const float4*)(lookup + (size_t)ent * DIM);
    float4*       dst = (float4*)(out + (size_t)i * DIM);
    dst[threadIdx.x] = src[threadIdx.x];
}

// Typed rows: out[row] = emb_t[didx] @ W_t + b_t via f32 WMMA.
// Block = 256 threads (8 waves). Block tile: 32 rows x 512 cols.
// Wave tile: 32 x 64 -> 8 accumulators; each B fragment feeds 2 WMMAs.
__global__ __launch_bounds__(256) void typed_gemm_kernel(
    const float* __restrict__ emb0, const float* __restrict__ emb1, const float* __restrict__ emb2,
    const float* __restrict__ W0,   const float* __restrict__ W1,   const float* __restrict__ W2,
    const float* __restrict__ b0,   const float* __restrict__ b1,   const float* __restrict__ b2,
    const int* __restrict__ counts, const int* __restrict__ rowlist, const int* __restrict__ didxlist,
    float* __restrict__ out, float* __restrict__ dummy)
{
    const int t     = blockIdx.z;
    const int count = counts[t];
    const int tile  = blockIdx.x;
    if (tile * ROWS >= count) return;   // uniform exit: survivors keep EXEC all-1s

    const float* emb = (t == 0) ? emb0 : ((t == 1) ? emb1 : emb2);
    const float* W   = (t == 0) ? W0   : ((t == 1) ? W1   : W2);
    const float* bs  = (t == 0) ? b0   : ((t == 1) ? b1   : b2);

    __shared__ float As[ROWS * LDSS];   // 32 x 384 f32 chunk, padded
    __shared__ int   sRow[ROWS];
    __shared__ int   sDidx[ROWS];

    if (threadIdx.x < ROWS) {
        int g = tile * ROWS + threadIdx.x;
        bool v = g < count;
        sRow[threadIdx.x]  = v ? rowlist[t * BATCH + g]  : -1;
        sDidx[threadIdx.x] = v ? didxlist[t * BATCH + g] : 0;
    }
    __syncthreads();

    const int lane = threadIdx.x & 31;
    const int wv   = threadIdx.x >> 5;
    const int h    = lane >> 4;       // lane-half: selects K pair (A) / M+8 (C)
    const int l15  = lane & 15;
    const int ncol = blockIdx.y * 512 + wv * 64 + l15;

    // Bias depends only on the column -> preload it into the accumulators.
    v8f acc[2][4];
#pragma unroll
    for (int s = 0; s < 4; ++s) {
        float bv = bs[ncol + s * 16];
        acc[0][s] = (v8f){bv, bv, bv, bv, bv, bv, bv, bv};
        acc[1][s] = acc[0][s];
    }

#pragma unroll
    for (int chunk = 0; chunk < 2; ++chunk) {
        // Stage 32 gathered rows x 384 cols (48KB) via b128: 3072 float4 / 256 thr.
#pragma unroll
        for (int it = 0; it < 12; ++it) {
            int g   = threadIdx.x + it * 256;   // 0..3071
            int row = g / 96;
            int c4  = g - row * 96;
            float4 v = *(const float4*)(emb + (size_t)sDidx[row] * IN_DIM
                                        + chunk * KCHUNK + c4 * 4);
            *(float4*)(&As[row * LDSS + c4 * 4]) = v;
        }
        __syncthreads();

        // A frag (16x4 f32): lane l15 = row M; VGPR pair holds K = k+2h, k+2h+1.
        const float* a0p = &As[l15 * LDSS + 2 * h];          // rows 0..15
        const float* a1p = &As[(16 + l15) * LDSS + 2 * h];   // rows 16..31
        const float* Wb  = W + (size_t)(chunk * KCHUNK + 2 * h) * DIM + ncol;

        for (int k = 0; k < KCHUNK; k += 4) {
            v2f a0 = *(const v2f*)(a0p + k);                 // ds_load_b64
            v2f a1 = *(const v2f*)(a1p + k);
            const float* wp = Wb + (size_t)k * DIM;
#pragma unroll
            for (int s = 0; s < 4; ++s) {
                v2f bf;
                bf.x = wp[s * 16];          // W[k+2h  ][n]
                bf.y = wp[s * 16 + DIM];    // W[k+2h+1][n]
                acc[0][s] = __builtin_amdgcn_wmma_f32_16x16x4_f32(
                    false, a0, false, bf, (short)0, acc[0][s], false, false);
                acc[1][s] = __builtin_amdgcn_wmma_f32_16x16x4_f32(
                    false, a1, false, bf, (short)0, acc[1][s], false, false);
            }
        }
        __syncthreads();   // protect As before next chunk's staging
    }

    // Epilogue: C/D layout VGPR r -> M = r (lanes 0-15) / r+8 (lanes 16-31).
    // Preload row ids; redirect padded rows to a dummy scratch row (branch-free).
    int rids[2][8];
#pragma unroll
    for (int mt = 0; mt < 2; ++mt)
#pragma unroll
        for (int r = 0; r < 8; ++r)
            rids[mt][r] = sRow[mt * 16 + 8 * h + r];

#pragma unroll
    for (int mt = 0; mt < 2; ++mt) {
#pragma unroll
        for (int r = 0; r < 8; ++r) {
            int rid = rids[mt][r];
            float* base = (rid >= 0) ? (out + (size_t)rid * DIM + ncol)
                                     : (dummy + l15);
#pragma unroll
            for (int s = 0; s < 4; ++s)
                base[s * 16] = acc[mt][s][r];
        }
    }
}

extern "C" void kernel_launch(void* const* d_in, const int* in_sizes, int n_in,
                              void* d_out, int out_size, void* d_ws, size_t ws_size,
                              hipStream_t stream) {
    const int*   indices = (const int*)d_in[0];
    const int*   etypes  = (const int*)d_in[1];
    const int*   edidx   = (const int*)d_in[2];
    const float* emb0    = (const float*)d_in[3];
    const float* emb1    = (const float*)d_in[4];
    const float* emb2    = (const float*)d_in[5];
    const float* W0      = (const float*)d_in[6];
    const float* W1      = (const float*)d_in[7];
    const float* W2      = (const float*)d_in[8];
    const float* b0      = (const float*)d_in[9];
    const float* b1      = (const float*)d_in[10];
    const float* b2      = (const float*)d_in[11];
    const float* lookup  = (const float*)d_in[12];
    float* out = (float*)d_out;

    // Workspace (ints): [counts pad 64][rowlist 4*B][didxlist 3*B][dummy row] ~463KB
    int*   counts   = (int*)d_ws;
    int*   rowlist  = counts + 64;
    int*   didxlist = rowlist + 4 * BATCH;
    float* dummy    = (float*)(didxlist + 3 * BATCH);

    zero_counts_kernel<<<1, 32, 0, stream>>>(counts);
    route_kernel<<<BATCH / 256, 256, 0, stream>>>(indices, etypes, edidx,
                                                  counts, rowlist, didxlist);
    lookup_copy_kernel<<<BATCH, 256, 0, stream>>>(counts, rowlist, indices, lookup, out);
    typed_gemm_kernel<<<dim3(BATCH / ROWS, 2, 3), 256, 0, stream>>>(
        emb0, emb1, emb2, W0, W1, W2, b0, b1, b2,
        counts, rowlist, didxlist, out, dummy);
}